// TrackAssocWrapper_49409303773743
// MI455X (gfx1250) — compile-verified
//
#include <hip/hip_runtime.h>

typedef __attribute__((ext_vector_type(2))) float v2f;
typedef __attribute__((ext_vector_type(8))) float v8f;

#define N_DETS 1024
#define N_TRKS 1024
#define DIM    128
#define LDT    36    // 32-float k-chunk + 4 pad floats -> conflict-free b64 frags

// ---- CDNA5 async global->LDS staging (guarded; falls back to ld/st) -------
#if defined(__gfx1250__) && __has_builtin(__builtin_amdgcn_global_load_async_to_lds_b128)
#define HAVE_ASYNC_LDS 1
#define ASG __attribute__((address_space(1)))
#define ASL __attribute__((address_space(3)))
typedef int b128_t __attribute__((vector_size(16)));   // matches builtin's int4 param
#endif
#ifndef HAVE_ASYNC_LDS
#define HAVE_ASYNC_LDS 0
#endif

static __device__ __forceinline__ void stage_b128(const float* __restrict__ g,
                                                  float* l) {
#if HAVE_ASYNC_LDS
    __builtin_amdgcn_global_load_async_to_lds_b128(
        (ASG b128_t*)g, (ASL b128_t*)l, 0, 0);
#else
    *(float4*)l = *(const float4*)g;
#endif
}

static __device__ __forceinline__ void stage_wait_barrier() {
#if HAVE_ASYNC_LDS
#if __has_builtin(__builtin_amdgcn_s_wait_asynccnt)
    __builtin_amdgcn_s_wait_asynccnt(0);
#else
    asm volatile("s_wait_asynccnt 0x0" ::: "memory");
#endif
#endif
    __syncthreads();
}

// ---------------------------------------------------------------------------
// Kernel 1: C[128 x 128] = A[rows, :] @ W + bias   (A row tile per block)
// W is staged TRANSPOSED in LDS so the B-fragment is one contiguous b64 load
// (kills the dword-repacking v_dual_mov traffic seen in round 0's asm).
// ---------------------------------------------------------------------------
__global__ __launch_bounds__(256) void linear128_kernel(
    const float* __restrict__ A, const float* __restrict__ W,
    const float* __restrict__ bias, float* __restrict__ C)
{
    const int tid  = threadIdx.x;
    const int lane = tid & 31;
    const int wave = tid >> 5;            // 0..7 -> 16-row strip
    const int n0   = blockIdx.x * 128;

    __shared__ __align__(16) float s_a [128 * LDT];   // rows  x k-chunk
    __shared__ __align__(16) float s_wT[128 * LDT];   // cols j x k-chunk (transposed)

    v8f acc[8] = {};
    const int fr = lane & 15;             // fragment row (M for A, N for B)
    const int kh = (lane >> 4) << 1;      // k sub-offset: 0 or 2

    for (int kk = 0; kk < DIM; kk += 32) {
        __syncthreads();                  // LDS reuse guard
        for (int i = tid; i < 128 * 8; i += 256) {            // A tile: 128x32
            int r = i >> 3, q = (i & 7) << 2;
            stage_b128(A + (size_t)(n0 + r) * DIM + kk + q, &s_a[r * LDT + q]);
        }
        for (int i = tid; i < 32 * 32; i += 256) {            // W tile: 32x128, transpose
            int r = i >> 5, q = (i & 31) << 2;                // r = k row, q = j col
            float4 v = *(const float4*)(W + (size_t)(kk + r) * DIM + q);
            s_wT[(q + 0) * LDT + r] = v.x;
            s_wT[(q + 1) * LDT + r] = v.y;
            s_wT[(q + 2) * LDT + r] = v.z;
            s_wT[(q + 3) * LDT + r] = v.w;
        }
        stage_wait_barrier();

#pragma unroll
        for (int k = 0; k < 32; k += 4) {
            v2f a = *(const v2f*)&s_a[(wave * 16 + fr) * LDT + k + kh];
#pragma unroll
            for (int t = 0; t < 8; ++t) {
                // B[k][j] = W[k][j]; transposed LDS makes (k,k+1) contiguous
                v2f b = *(const v2f*)&s_wT[(t * 16 + fr) * LDT + k + kh];
                acc[t] = __builtin_amdgcn_wmma_f32_16x16x4_f32(
                    false, a, false, b, (short)0, acc[t], false, false);
            }
        }
    }

    const int rowv = wave * 16 + ((lane >> 4) << 3);   // + v gives row in tile
#pragma unroll
    for (int t = 0; t < 8; ++t) {
        int j = t * 16 + fr;
        float bj = bias[j];
#pragma unroll
        for (int v = 0; v < 8; ++v) {
            int n = n0 + rowv + v;
            C[(size_t)n * DIM + j] = acc[t][v] + bj;
        }
    }
}

// ---------------------------------------------------------------------------
// Kernel 2: out[n,m] = (obj[n,:]·trk[m,:] + rel[n,m,:]·wsum + bsum) / sqrt(D)
// Block tile 128 (n) x 64 (m); 8 waves x (16 rows x 64 cols) = 4 WMMA accums.
// rel_dist (the dominant 12.6 MB stream) is read exactly once, in epilogue.
// ---------------------------------------------------------------------------
__global__ __launch_bounds__(256) void fuse_gemm_kernel(
    const float* __restrict__ obj, const float* __restrict__ trk,
    const float* __restrict__ rel, const float* __restrict__ W_rel,
    const float* __restrict__ b_rel, float* __restrict__ out)
{
    const int tid  = threadIdx.x;
    const int lane = tid & 31;
    const int wave = tid >> 5;
    const int n0   = blockIdx.x * 128;    // detection rows
    const int m0   = blockIdx.y * 64;     // track cols

    __shared__ __align__(16) float s_obj[128 * LDT];
    __shared__ __align__(16) float s_trk[64 * LDT];

    // Uniform (scalar-load) reduction: row-sums of W_rel and sum of b_rel.
    float w0 = 0.f, w1 = 0.f, w2 = 0.f, bs = 0.f;
    for (int d = 0; d < DIM; ++d) {
        w0 += W_rel[d];
        w1 += W_rel[DIM + d];
        w2 += W_rel[2 * DIM + d];
        bs += b_rel[d];
    }

    v8f acc[4] = {};
    const int fr = lane & 15;
    const int kh = (lane >> 4) << 1;

    for (int kk = 0; kk < DIM; kk += 32) {
        __syncthreads();                  // LDS reuse guard
        for (int i = tid; i < 128 * 8; i += 256) {            // obj tile 128x32
            int r = i >> 3, q = (i & 7) << 2;
            stage_b128(obj + (size_t)(n0 + r) * DIM + kk + q, &s_obj[r * LDT + q]);
        }
        for (int i = tid; i < 64 * 8; i += 256) {             // trk tile 64x32
            int r = i >> 3, q = (i & 7) << 2;
            stage_b128(trk + (size_t)(m0 + r) * DIM + kk + q, &s_trk[r * LDT + q]);
        }
        stage_wait_barrier();

#pragma unroll
        for (int k = 0; k < 32; k += 4) {
            v2f a = *(const v2f*)&s_obj[(wave * 16 + fr) * LDT + k + kh];
#pragma unroll
            for (int t = 0; t < 4; ++t) {
                // B = trk^T : B[k][n] = trk[n][k] -> two k's are contiguous
                v2f b = *(const v2f*)&s_trk[(t * 16 + fr) * LDT + k + kh];
                acc[t] = __builtin_amdgcn_wmma_f32_16x16x4_f32(
                    false, a, false, b, (short)0, acc[t], false, false);
            }
        }
    }

    const float inv  = 0.088388347648318447f;   // 1/sqrt(128)
    const int   rowv = wave * 16 + ((lane >> 4) << 3);
#pragma unroll
    for (int t = 0; t < 4; ++t) {
        int m = m0 + t * 16 + fr;
#pragma unroll
        for (int v = 0; v < 8; ++v) {
            int n = n0 + rowv + v;
            const float* rp = rel + ((size_t)n * N_TRKS + m) * 3;
            float s = acc[t][v] + rp[0] * w0 + rp[1] * w1 + rp[2] * w2 + bs;
            out[(size_t)n * N_TRKS + m] = s * inv;
        }
    }
}

// ---------------------------------------------------------------------------
extern "C" void kernel_launch(void* const* d_in, const int* in_sizes, int n_in,
                              void* d_out, int out_size, void* d_ws, size_t ws_size,
                              hipStream_t stream)
{
    (void)in_sizes; (void)n_in; (void)out_size; (void)ws_size;

    const float* det   = (const float*)d_in[0];  // [1024,128]
    const float* trk   = (const float*)d_in[1];  // [1024,128]
    const float* rel   = (const float*)d_in[2];  // [1024,1024,3]
    const float* W_det = (const float*)d_in[3];  // [128,128]
    const float* b_det = (const float*)d_in[4];  // [128]
    const float* W_trk = (const float*)d_in[5];  // [128,128]
    const float* b_trk = (const float*)d_in[6];  // [128]
    const float* W_rel = (const float*)d_in[7];  // [3,128]
    const float* b_rel = (const float*)d_in[8];  // [128]

    float* out_mat = (float*)d_out;                          // [1024,1024]
    float* out_trk = out_mat + (size_t)N_DETS * N_TRKS;      // [1024,128]
    float* obj_ws  = (float*)d_ws;                           // [1024,128] scratch

    dim3 blk(256);
    // obj = det @ W_det + b_det   (scratch, consumed by fuse_gemm)
    linear128_kernel<<<dim3(N_DETS / 128), blk, 0, stream>>>(det, W_det, b_det, obj_ws);
    // second output: trk_nb = det @ W_trk + b_trk
    linear128_kernel<<<dim3(N_DETS / 128), blk, 0, stream>>>(det, W_trk, b_trk, out_trk);
    // main output
    fuse_gemm_kernel<<<dim3(N_DETS / 128, N_TRKS / 64), blk, 0, stream>>>(
        obj_ws, trk, rel, W_rel, b_rel, out_mat);
}